// GraphAttentionLayer_27358941675624
// MI455X (gfx1250) — compile-verified
//
#include <hip/hip_runtime.h>
#include <hip/hip_bf16.h>
#include <math.h>

// ---------------------------------------------------------------------------
// Graph attention layer, MI455X (gfx1250), wave32 + V_WMMA_F32_16X16X4_F32.
// N=1024 nodes, D=128, H=4 heads, HD=32, E=16 edge features.
//
// Restructured math (avoids the 512MB ef tensor):
//   A[i,h,e]  = sum_d q[i,h,d] * We[e, h*32+d]
//   bb[i,h]   = sum_d q[i,h,d] * be[h*32+d]
//   score[i,j,h] = scale * ( q_i . k_j  +  sum_e edges[i,j,e]*A[i,h,e] + bb[i,h] )
// One workgroup per 16-row i-tile keeps the full [4][16][1024] score slab in
// CDNA5's 320KB LDS so edges stream through HBM exactly once (~64MB total).
// ---------------------------------------------------------------------------

typedef __attribute__((ext_vector_type(2))) float v2f;
typedef __attribute__((ext_vector_type(4))) float v4f;
typedef __attribute__((ext_vector_type(8))) float v8f;

__device__ __forceinline__ v8f wmma4(v2f a, v2f b, v8f c) {
  // D = A(16x4, f32) * B(4x16, f32) + C(16x16, f32)
  return __builtin_amdgcn_wmma_f32_16x16x4_f32(
      /*neg_a=*/false, a, /*neg_b=*/false, b,
      /*c_mod=*/(short)0, c, /*reuse_a=*/false, /*reuse_b=*/false);
}

#define NND 1024
#define DDIM 128
#define EDIM 16

// ---------------------------------------------------------------------------
// Kernel 1: q/k/v projections.  grid=(64,6), block=128 (4 waves).
// Each wave computes one 16x16 output tile, K=128 -> 32 WMMAs.
// ---------------------------------------------------------------------------
__global__ __launch_bounds__(128, 1) void gat_qkv_kernel(
    const float* __restrict__ nodes,
    const float* __restrict__ Wq, const float* __restrict__ bq,
    const float* __restrict__ Wk, const float* __restrict__ bk,
    const float* __restrict__ Wv, const float* __restrict__ bv,
    float* __restrict__ qo, float* __restrict__ ko, float* __restrict__ vo) {
  const int t = threadIdx.x, wv = t >> 5, lane = t & 31;
  const int nlo = lane & 15;
  const int koff = (lane < 16) ? 0 : 2;
  const int mhi  = (lane < 16) ? 0 : 8;
  const int ct = blockIdx.y * 4 + wv;       // 0..23
  const int p = ct >> 3;                    // 0:q 1:k 2:v
  const int c0 = (ct & 7) * 16;
  const int i0 = blockIdx.x * 16;
  const float* W    = (p == 0) ? Wq : (p == 1) ? Wk : Wv;
  const float* bias = (p == 0) ? bq : (p == 1) ? bk : bv;
  float* outp       = (p == 0) ? qo : (p == 1) ? ko : vo;

  v8f C = {};
  const float* arow = nodes + (size_t)(i0 + nlo) * DDIM + koff;
  for (int d0 = 0; d0 < DDIM; d0 += 4) {
    v2f a = *(const v2f*)(arow + d0);                       // K=d0+koff, +1
    v2f b;
    b.x = W[(size_t)(d0 + koff) * DDIM + c0 + nlo];
    b.y = W[(size_t)(d0 + koff + 1) * DDIM + c0 + nlo];
    C = wmma4(a, b, C);
  }
#pragma unroll
  for (int r = 0; r < 8; ++r) {
    const int il = r + mhi, c = c0 + nlo;
    outp[(size_t)(i0 + il) * DDIM + c] = C[r] + bias[c];
  }
}

// ---------------------------------------------------------------------------
// Kernel 2: edge-projection coefficients A[i,h,e] and bb[i,h].  Tiny.
// grid=256, block=256 : 4 nodes per block, thread r=(h,e) does 32 MACs.
// ---------------------------------------------------------------------------
__global__ __launch_bounds__(256, 1) void gat_coef_kernel(
    const float* __restrict__ q, const float* __restrict__ We,
    const float* __restrict__ be, float* __restrict__ Ao,
    float* __restrict__ bbo) {
  const int t = threadIdx.x;
  const int i = blockIdx.x * 4 + (t >> 6);
  const int r = t & 63, h = r >> 4, e = r & 15;
  const float* qp = q + (size_t)i * DDIM + h * 32;
  const float* wp = We + (size_t)e * DDIM + h * 32;
  float s = 0.f;
#pragma unroll
  for (int d = 0; d < 32; ++d) s += qp[d] * wp[d];
  Ao[(size_t)i * 64 + r] = s;
  if (e == 0) {
    const float* bp = be + h * 32;
    float sb = 0.f;
#pragma unroll
    for (int d = 0; d < 32; ++d) sb += qp[d] * bp[d];
    bbo[(size_t)i * 4 + h] = sb;
  }
}

// ---------------------------------------------------------------------------
// Kernel 3: attention + aggregation + output proj + residual + LayerNorm.
// One workgroup per 16-row i-tile; 8 waves: wave = (head h = w&3, group g=w>>2)
// where group g handles j-tiles with parity g.  Scores live in LDS (~310KB).
// ---------------------------------------------------------------------------
struct alignas(16) SmemB {
  float scores[4][16][NND];   // 262144 B : unnormalized p after softmax phase
  float qs[16][DDIM];         //   8192 B
  float As[16][4][EDIM];      //   4096 B
  float bbs[16][4];           //    256 B
  float redmax[64][4];        //   1024 B
  float redsum[64][4];        //   1024 B
  float rowinvl[64];          //    256 B
  float aggP[2][4][16][32];   //  16384 B : per-group partial attn@v
  float aggC[16][DDIM];       //   8192 B : combined agg (post 1/l)
  float preL[16][DDIM];       //   8192 B : pre-LayerNorm
  float mu[16];
  float rv[16];
};                            // total ~310 KB (< 320 KB CDNA5 LDS)

__global__ __launch_bounds__(256, 1) void gat_attn_kernel(
    const float* __restrict__ q, const float* __restrict__ k,
    const float* __restrict__ v, const float* __restrict__ Acoef,
    const float* __restrict__ bb, const float* __restrict__ edges,
    const int* __restrict__ adj, const float* __restrict__ nodes,
    const float* __restrict__ Wo, const float* __restrict__ bo,
    const float* __restrict__ ln_s, const float* __restrict__ ln_b,
    float* __restrict__ out, float* __restrict__ avg_attn) {
  __shared__ SmemB sm;
  const int t = threadIdx.x;
  const int wv = t >> 5, lane = t & 31;
  const int h = wv & 3, g = wv >> 2;
  const int i0 = blockIdx.x * 16;
  const int nlo = lane & 15;
  const int koff = (lane < 16) ? 0 : 2;
  const int mhi  = (lane < 16) ? 0 : 8;
  const int hb = h * 32;
  const float scale = 0.17677669529663687f;  // 1/sqrt(32)

  // Stage q tile, A coefficients, bias dots into LDS.
  for (int idx = t; idx < 16 * DDIM; idx += 256)
    sm.qs[idx >> 7][idx & 127] = q[(size_t)(i0 + (idx >> 7)) * DDIM + (idx & 127)];
  for (int idx = t; idx < 16 * 64; idx += 256) {
    const int il = idx >> 6, r = idx & 63;
    sm.As[il][r >> 4][r & 15] = Acoef[(size_t)(i0 + il) * 64 + r];
  }
  if (t < 64) sm.bbs[t >> 2][t & 3] = bb[(size_t)(i0 + (t >> 2)) * 4 + (t & 3)];
  __syncthreads();

  // ---- Pass 1: scores = scale*(q.k^T + edge-bias), masked, into LDS -------
  for (int jt = g; jt < 64; jt += 2) {
    const int j0 = jt * 16;
    v8f C = {};
    const float* kcol = k + (size_t)(j0 + nlo) * DDIM + hb + koff;
    const float* qrow = &sm.qs[nlo][hb + koff];
#pragma unroll
    for (int d0 = 0; d0 < 32; d0 += 4) {
      v2f a = *(const v2f*)(qrow + d0);
      v2f b = *(const v2f*)(kcol + d0);
      C = wmma4(a, b, C);
    }
    const int gj = j0 + nlo;
#pragma unroll
    for (int r = 0; r < 8; ++r) {
      const int il = r + mhi;
      const int gi = i0 + il;
      const float* ep = edges + ((size_t)gi * NND + gj) * EDIM;
      const float* as = &sm.As[il][h][0];
      float s2 = sm.bbs[il][h];
#pragma unroll
      for (int e4 = 0; e4 < EDIM; e4 += 4) {
        v4f ev = *(const v4f*)(ep + e4);
        s2 += ev.x * as[e4] + ev.y * as[e4 + 1] + ev.z * as[e4 + 2] + ev.w * as[e4 + 3];
      }
      float s = (C[r] + s2) * scale;
      if (adj[(size_t)gi * NND + gj] == 0) s = -1e9f;
      sm.scores[h][il][gj] = s;
    }
  }
  __syncthreads();

  // ---- Softmax over j: 64 rows (h,il), 4 threads per row ------------------
  {
    const int r = t >> 2, part = t & 3;
    const int hh = r >> 4, il = r & 15;
    float* row = sm.scores[hh][il];
    float m = -3.4e38f;
    for (int j = part * 256; j < part * 256 + 256; ++j) m = fmaxf(m, row[j]);
    sm.redmax[r][part] = m;
    __syncthreads();
    m = fmaxf(fmaxf(sm.redmax[r][0], sm.redmax[r][1]),
              fmaxf(sm.redmax[r][2], sm.redmax[r][3]));
    float ssum = 0.f;
    for (int j = part * 256; j < part * 256 + 256; ++j) {
      const float p = __expf(row[j] - m);
      row[j] = p;              // keep unnormalized; fold 1/l later
      ssum += p;
    }
    sm.redsum[r][part] = ssum;
    __syncthreads();
    if (part == 0) {
      const float l = sm.redsum[r][0] + sm.redsum[r][1] +
                      sm.redsum[r][2] + sm.redsum[r][3];
      sm.rowinvl[r] = 1.0f / l;
    }
  }
  __syncthreads();

  // ---- avg_attn[i,j] = mean_h p/l ----------------------------------------
  for (int idx = t; idx < 16 * NND; idx += 256) {
    const int il = idx >> 10, j = idx & 1023;
    float s = 0.f;
#pragma unroll
    for (int hh = 0; hh < 4; ++hh)
      s += sm.scores[hh][il][j] * sm.rowinvl[hh * 16 + il];
    avg_attn[(size_t)(i0 + il) * NND + j] = 0.25f * s;
  }

  // ---- Pass 2: agg[i,h,d] = sum_j p * v[j,h,d]  (WMMA, K=j) ---------------
#pragma unroll
  for (int dt = 0; dt < 2; ++dt) {
    const int d0 = dt * 16;
    v8f C = {};
    for (int jt = g; jt < 64; jt += 2) {
      const int j0 = jt * 16;
#pragma unroll
      for (int kk = 0; kk < 16; kk += 4) {
        const int kb = j0 + kk + koff;
        v2f a = *(const v2f*)&sm.scores[h][nlo][kb];        // A: rows=il, K=j
        v2f b;
        b.x = v[(size_t)kb * DDIM + hb + d0 + nlo];
        b.y = v[(size_t)(kb + 1) * DDIM + hb + d0 + nlo];
        C = wmma4(a, b, C);
      }
    }
#pragma unroll
    for (int r = 0; r < 8; ++r) sm.aggP[g][h][r + mhi][d0 + nlo] = C[r];
  }
  __syncthreads();

  // Combine the two j-parity groups and apply 1/l.
  for (int idx = t; idx < 16 * DDIM; idx += 256) {
    const int il = idx >> 7, d = idx & 127;
    const int hh = d >> 5;
    sm.aggC[il][d] = (sm.aggP[0][hh][il][d & 31] + sm.aggP[1][hh][il][d & 31]) *
                     sm.rowinvl[hh * 16 + il];
  }
  __syncthreads();

  // ---- Output projection agg@Wo + bo + residual (WMMA, wave = 16 cols) ----
  {
    const int c0 = wv * 16;
    v8f C = {};
    for (int d0 = 0; d0 < DDIM; d0 += 4) {
      v2f a = *(const v2f*)&sm.aggC[nlo][d0 + koff];
      v2f b;
      b.x = Wo[(size_t)(d0 + koff) * DDIM + c0 + nlo];
      b.y = Wo[(size_t)(d0 + koff + 1) * DDIM + c0 + nlo];
      C = wmma4(a, b, C);
    }
#pragma unroll
    for (int r = 0; r < 8; ++r) {
      const int il = r + mhi, c = c0 + nlo;
      sm.preL[il][c] = C[r] + bo[c] + nodes[(size_t)(i0 + il) * DDIM + c];
    }
  }
  __syncthreads();

  // ---- LayerNorm ----------------------------------------------------------
  if (t < 16) {
    float s = 0.f, s2 = 0.f;
    for (int c = 0; c < DDIM; ++c) {
      const float x = sm.preL[t][c];
      s += x;
      s2 += x * x;
    }
    const float m = s * (1.0f / 128.0f);
    sm.mu[t] = m;
    sm.rv[t] = rsqrtf(s2 * (1.0f / 128.0f) - m * m + 1e-6f);
  }
  __syncthreads();
  for (int idx = t; idx < 16 * DDIM; idx += 256) {
    const int il = idx >> 7, c = idx & 127;
    out[(size_t)(i0 + il) * DDIM + c] =
        (sm.preL[il][c] - sm.mu[il]) * sm.rv[il] * ln_s[c] + ln_b[c];
  }
}

// ---------------------------------------------------------------------------
extern "C" void kernel_launch(void* const* d_in, const int* in_sizes, int n_in,
                              void* d_out, int out_size, void* d_ws,
                              size_t ws_size, hipStream_t stream) {
  const float* nodes = (const float*)d_in[0];
  const float* edges = (const float*)d_in[1];
  const int*   adj   = (const int*)d_in[2];
  const float* Wq = (const float*)d_in[3];  const float* bq = (const float*)d_in[4];
  const float* Wk = (const float*)d_in[5];  const float* bk = (const float*)d_in[6];
  const float* Wv = (const float*)d_in[7];  const float* bv = (const float*)d_in[8];
  const float* We = (const float*)d_in[9];  const float* be = (const float*)d_in[10];
  const float* Wo = (const float*)d_in[11]; const float* bo = (const float*)d_in[12];
  const float* ln_s = (const float*)d_in[13];
  const float* ln_b = (const float*)d_in[14];

  float* ws = (float*)d_ws;
  float* qb  = ws;                 // 1024*128
  float* kb  = qb + NND * DDIM;    // 1024*128
  float* vb  = kb + NND * DDIM;    // 1024*128
  float* Ab  = vb + NND * DDIM;    // 1024*64
  float* bbv = Ab + NND * 64;      // 1024*4   (~1.8 MB total scratch)

  float* outp = (float*)d_out;                 // [1024,128]
  float* avgp = outp + (size_t)NND * DDIM;     // [1024,1024]

  gat_qkv_kernel<<<dim3(64, 6), 128, 0, stream>>>(nodes, Wq, bq, Wk, bk, Wv, bv,
                                                  qb, kb, vb);
  gat_coef_kernel<<<256, 256, 0, stream>>>(qb, We, be, Ab, bbv);
  gat_attn_kernel<<<64, 256, 0, stream>>>(qb, kb, vb, Ab, bbv, edges, adj,
                                          nodes, Wo, bo, ln_s, ln_b, outp, avgp);
}